// SubsetOperator_16106127360458
// MI455X (gfx1250) — compile-verified
//
#include <hip/hip_runtime.h>
#include <math.h>

// Problem constants (match reference: B=4096, N=8192, K=8, TAU=1, HARD=False)
#define B_ROWS  4096
#define N_COLS  8192
#define K_ITERS 8
#define TPB     1024                 // 32 wave32 waves per workgroup
#define EPT     (N_COLS / TPB)       // 8 elements per thread
#define EPSILON 1.175494350822287508e-38f   // np.finfo(float32).tiny

static_assert(EPT == 8, "layout assumes 8 elements per thread");

// ---- CDNA5 async global->LDS DMA (ASYNCcnt path), guarded so we always compile ----
#if defined(__HIP_DEVICE_COMPILE__) && __has_builtin(__builtin_amdgcn_global_load_async_to_lds_b128)
#define HAVE_ASYNC 1
#else
#define HAVE_ASYNC 0
#endif

#if HAVE_ASYNC
typedef int v4i_t __attribute__((vector_size(16)));                 // b128 payload type
typedef __attribute__((address_space(1))) v4i_t* gptr_b128_t;       // global src
typedef __attribute__((address_space(3))) v4i_t* lptr_b128_t;       // LDS dst
#endif

__device__ __forceinline__ void wait_async_then_barrier() {
#if HAVE_ASYNC
#if __has_builtin(__builtin_amdgcn_s_wait_asynccnt)
    __builtin_amdgcn_s_wait_asynccnt(0);
#else
    asm volatile("s_wait_asynccnt 0" ::: "memory");
#endif
#endif
    __syncthreads();
}

// Block-wide reduction over 1024 threads = 32 waves of 32 lanes.
// Stage 1: shfl_xor tree inside each wave. Stage 2: wave 0 (exactly 32 lanes)
// reduces the 32 per-wave partials. Leading barrier protects red[] against
// the previous round's readers (write-after-read across calls).
template <bool IS_MAX>
__device__ __forceinline__ float block_reduce(float v, float* red, int wave, int lane) {
#pragma unroll
    for (int m = 16; m >= 1; m >>= 1) {
        float o = __shfl_xor(v, m, 32);
        v = IS_MAX ? fmaxf(v, o) : (v + o);
    }
    __syncthreads();
    if (lane == 0) red[wave] = v;
    __syncthreads();
    if (wave == 0) {
        float t = red[lane];
#pragma unroll
        for (int m = 16; m >= 1; m >>= 1) {
            float o = __shfl_xor(t, m, 32);
            t = IS_MAX ? fmaxf(t, o) : (t + o);
        }
        if (lane == 0) red[32] = t;
    }
    __syncthreads();
    return red[32];
}

__global__ __launch_bounds__(TPB) void subset_topk_kernel(
    const float* __restrict__ scores,
    const float* __restrict__ gnoise,
    float* __restrict__ out)
{
    __shared__ float sm[N_COLS];   // staged scores row (32 KB) via async DMA
    __shared__ float red[34];      // reduction scratch

    const int       tid  = threadIdx.x;
    const int       wave = tid >> 5;
    const int       lane = tid & 31;
    const long long row  = blockIdx.x;

    const float* srow = scores + row * (long long)N_COLS;
    const float* grow = gnoise + row * (long long)N_COLS;
    const int    base = tid * EPT;

#if HAVE_ASYNC
    {
        // DMA scores row into LDS: 2 x b128 per lane (imm offset is added to
        // both the global and LDS addresses per the ISA), 1024 lanes -> 32 KB.
        const char* src = (const char*)srow + tid * 16;
        char*       dst = (char*)sm + tid * 16;
        __builtin_amdgcn_global_load_async_to_lds_b128(
            (gptr_b128_t)src, (lptr_b128_t)dst, 0, 0);
        __builtin_amdgcn_global_load_async_to_lds_b128(
            (gptr_b128_t)src, (lptr_b128_t)dst, 16384, 0);
    }
#else
    {
        float4*       dst4 = (float4*)sm + tid * 2;
        const float4* s4   = (const float4*)srow + tid * 2;
        dst4[0] = s4[0];
        dst4[1] = s4[1];
    }
#endif

    // Gumbel noise row straight to VGPRs; overlaps with the async DMA above.
    float gv[EPT];
    {
        const float4* g4 = (const float4*)grow + tid * 2;
        float4 ga = g4[0], gb = g4[1];
        gv[0] = ga.x; gv[1] = ga.y; gv[2] = ga.z; gv[3] = ga.w;
        gv[4] = gb.x; gv[5] = gb.y; gv[6] = gb.z; gv[7] = gb.w;
    }

    wait_async_then_barrier();

    // s0 = scores + g ; row max for the (single) softmax shift
    float s[EPT];
    float lmax = -INFINITY;
#pragma unroll
    for (int i = 0; i < EPT; ++i) {
        float v = sm[base + i] + gv[i];
        s[i] = v;
        lmax = fmaxf(lmax, v);
    }
    const float c = block_reduce<true>(lmax, red, wave, lane);

    // Reformulation: e = exp(s0 - c) computed ONCE; running multiplicative
    // mask M_k = prod_j max(1 - onehot_j, eps) replaces s += log(mask).
    // softmax(s_k) == M*e / sum(M*e) exactly (TAU == 1).
    float e[EPT], M[EPT], kh[EPT];
#pragma unroll
    for (int i = 0; i < EPT; ++i) {
        e[i]  = __expf(s[i] - c);
        M[i]  = 1.0f;
        kh[i] = 0.0f;
    }

#pragma unroll
    for (int k = 0; k < K_ITERS; ++k) {
        float w[EPT];
        float part = 0.0f;
#pragma unroll
        for (int i = 0; i < EPT; ++i) {
            w[i] = M[i] * e[i];
            part += w[i];
        }
        const float S   = block_reduce<false>(part, red, wave, lane);
        const float inv = 1.0f / S;
#pragma unroll
        for (int i = 0; i < EPT; ++i) {
            const float oh = w[i] * inv;   // onehot_k
            kh[i] += oh;
            M[i] *= fmaxf(1.0f - oh, EPSILON);
        }
    }

    // Coalesced output: 2 x b128 per lane
    float4* o4 = (float4*)(out + row * (long long)N_COLS) + tid * 2;
    o4[0] = make_float4(kh[0], kh[1], kh[2], kh[3]);
    o4[1] = make_float4(kh[4], kh[5], kh[6], kh[7]);
}

extern "C" void kernel_launch(void* const* d_in, const int* in_sizes, int n_in,
                              void* d_out, int out_size, void* d_ws, size_t ws_size,
                              hipStream_t stream) {
    (void)in_sizes; (void)n_in; (void)d_ws; (void)ws_size; (void)out_size;
    const float* scores = (const float*)d_in[0];
    const float* g      = (const float*)d_in[1];
    float*       out    = (float*)d_out;
    subset_topk_kernel<<<dim3(B_ROWS), dim3(TPB), 0, stream>>>(scores, g, out);
}